// TA_GCN_13331578486893
// MI455X (gfx1250) — compile-verified
//
#include <hip/hip_runtime.h>
#include <math.h>

#define N_NODES 100000
#define N_EDGES 1600000
#define IN_F    11
#define HID_F   128
#define OUT_F   2

typedef __attribute__((ext_vector_type(2))) float v2f;
typedef __attribute__((ext_vector_type(8))) float v8f;

// ---- order-preserving float <-> uint encoding for atomic min ----
__device__ __forceinline__ unsigned ord_enc(float v) {
    unsigned u = __float_as_uint(v);
    return (u & 0x80000000u) ? ~u : (u | 0x80000000u);
}
__device__ __forceinline__ float ord_dec(unsigned e) {
    unsigned u = (e & 0x80000000u) ? (e & 0x7FFFFFFFu) : ~e;
    return __uint_as_float(u);
}

// ---- elementwise init ----
__global__ void k_init_u32(unsigned* __restrict__ p, unsigned v, unsigned n) {
    unsigned i = blockIdx.x * 256u + threadIdx.x;
    if (i < n) p[i] = v;
}

// ---- in-degree ----
__global__ void k_degree(const int* __restrict__ col, unsigned* __restrict__ deg) {
    unsigned e = blockIdx.x * 256u + threadIdx.x;
    if (e < N_EDGES) atomicAdd(&deg[col[e]], 1u);
}

__global__ void k_dinv(const unsigned* __restrict__ deg, float* __restrict__ dinv) {
    unsigned i = blockIdx.x * 256u + threadIdx.x;
    if (i < N_NODES) {
        unsigned d = deg[i];
        dinv[i] = d ? rsqrtf((float)d) : 0.0f;
    }
}

// ---- one min-aggregation hop: enc[col][f] = min(enc, ord(norm * src[row][f])) ----
__global__ void k_hop(const float* __restrict__ src, const int* __restrict__ row,
                      const int* __restrict__ col, const float* __restrict__ dinv,
                      unsigned* __restrict__ enc, int F, int logFpad) {
    unsigned t = blockIdx.x * 256u + threadIdx.x;
    unsigned e = t >> logFpad;
    unsigned f = t & ((1u << logFpad) - 1u);
    if (e < N_EDGES && f < (unsigned)F) {
        int r = col ? row[e] : 0;
        int c = col[e];
        float nrm = dinv[r] * dinv[c];
        float v = nrm * src[(size_t)r * (unsigned)F + f];
        atomicMin(&enc[(size_t)c * (unsigned)F + f], ord_enc(v));
    }
}

// ---- decode encoded mins in place; empty segments (deg==0) -> 0 ----
__global__ void k_decode(unsigned* __restrict__ buf, const unsigned* __restrict__ deg,
                         int F, unsigned total) {
    unsigned t = blockIdx.x * 256u + threadIdx.x;
    if (t < total) {
        unsigned node = t / (unsigned)F;
        float v = deg[node] ? ord_dec(buf[t]) : 0.0f;
        ((float*)buf)[t] = v;
    }
}

// ---- build packed weight panels: Wc1 [36 x 128] (K-pad 11->12 per hop), Wc2 [384 x 16] (2 valid cols) ----
__global__ void k_wprep(const float* __restrict__ w1, const float* __restrict__ w2,
                        float* __restrict__ Wc1, float* __restrict__ Wc2) {
    int i = blockIdx.x * 256 + threadIdx.x;
    if (i < 36 * 128) {
        int r = i >> 7, c = i & 127;
        int seg = r / 12, off = r % 12;
        Wc1[i] = (off < IN_F) ? w1[(size_t)(seg * IN_F + off) * HID_F + c] : 0.0f;
    }
    if (i < 384 * 16) {
        int r = i >> 4, c = i & 15;
        int seg = r >> 7, off = r & 127;
        Wc2[i] = (c < OUT_F) ? w2[(size_t)(seg * HID_F + off) * OUT_F + c] : 0.0f;
    }
}

// ---- layer 1: hid = relu([x | h1 | h2] @ Wc1 + b1), fp32 WMMA 16x16x4 ----
// grid 3125 x 64 threads (2 waves, tiles = 2*blockIdx + waveId, exactly 6250 tiles)
__global__ __launch_bounds__(64) void k_gemm1(const float* __restrict__ x,
                                              const float* __restrict__ h1,
                                              const float* __restrict__ h2,
                                              const float* __restrict__ Wc1,
                                              const float* __restrict__ b1,
                                              float* __restrict__ hid) {
    __shared__ float Wl[36 * 128];
    for (int i = threadIdx.x; i < 36 * 128; i += 64) Wl[i] = Wc1[i];
    __syncthreads();

    unsigned lane = threadIdx.x & 31u;
    unsigned tile = blockIdx.x * 2u + (threadIdx.x >> 5);
    unsigned m = lane & 15u, hi = lane >> 4;

    v8f acc[8];
#pragma unroll
    for (int t = 0; t < 8; t++) acc[t] = (v8f){0.f, 0.f, 0.f, 0.f, 0.f, 0.f, 0.f, 0.f};

    size_t node = (size_t)tile * 16 + m;
#pragma unroll
    for (int ks = 0; ks < 9; ks++) {
        unsigned kk = (unsigned)ks * 4u + hi * 2u;
        unsigned seg = kk / 12u, off = kk % 12u;
        const float* s = (seg == 0u) ? x : ((seg == 1u) ? h1 : h2);
        v2f a;
        a.x = s[node * IN_F + off];
        a.y = (off < IN_F - 1u) ? s[node * IN_F + off + 1u] : 0.0f;
#pragma unroll
        for (int t = 0; t < 8; t++) {
            unsigned n = (unsigned)t * 16u + m;
            v2f b;
            b.x = Wl[kk * 128u + n];
            b.y = Wl[(kk + 1u) * 128u + n];
            acc[t] = __builtin_amdgcn_wmma_f32_16x16x4_f32(
                false, a, false, b, (short)0, acc[t], false, false);
        }
    }

#pragma unroll
    for (int t = 0; t < 8; t++) {
#pragma unroll
        for (int j = 0; j < 8; j++) {
            unsigned mr = (unsigned)j + 8u * hi;
            size_t nd = (size_t)tile * 16 + mr;
            unsigned nc = (unsigned)t * 16u + m;
            float v = acc[t][j] + b1[nc];
            hid[nd * HID_F + nc] = fmaxf(v, 0.0f);
        }
    }
}

// ---- layer 2: logits = [hid | g1 | g2] @ Wc2 + b2, K = 384, fp32 WMMA ----
__global__ __launch_bounds__(64) void k_gemm2(const float* __restrict__ h,
                                              const float* __restrict__ g1,
                                              const float* __restrict__ g2,
                                              const float* __restrict__ Wc2,
                                              const float* __restrict__ b2,
                                              float* __restrict__ out) {
    __shared__ float Wl[384 * 16];
    for (int i = threadIdx.x; i < 384 * 16; i += 64) Wl[i] = Wc2[i];
    __syncthreads();

    unsigned lane = threadIdx.x & 31u;
    unsigned tile = blockIdx.x * 2u + (threadIdx.x >> 5);
    unsigned m = lane & 15u, hi = lane >> 4;

    v8f acc = (v8f){0.f, 0.f, 0.f, 0.f, 0.f, 0.f, 0.f, 0.f};
    size_t node = (size_t)tile * 16 + m;
#pragma unroll
    for (int ks = 0; ks < 96; ks++) {
        unsigned kk = (unsigned)ks * 4u + hi * 2u;
        unsigned seg = kk >> 7, off = kk & 127u;
        const float* s = (seg == 0u) ? h : ((seg == 1u) ? g1 : g2);
        v2f a;
        a.x = s[node * HID_F + off];
        a.y = s[node * HID_F + off + 1u];
        v2f b;
        b.x = Wl[kk * 16u + m];
        b.y = Wl[(kk + 1u) * 16u + m];
        acc = __builtin_amdgcn_wmma_f32_16x16x4_f32(
            false, a, false, b, (short)0, acc, false, false);
    }

#pragma unroll
    for (int j = 0; j < 8; j++) {
        unsigned mr = (unsigned)j + 8u * hi;
        size_t nd = (size_t)tile * 16 + mr;
        if (m < OUT_F) out[nd * OUT_F + m] = acc[j] + b2[m];
    }
}

// ---- 2-class log_softmax in place on d_out ----
__global__ void k_lsm(float* __restrict__ out) {
    unsigned i = blockIdx.x * 256u + threadIdx.x;
    if (i < N_NODES) {
        float a = out[2 * i], b = out[2 * i + 1];
        float mx = fmaxf(a, b);
        float lse = mx + logf(expf(a - mx) + expf(b - mx));
        out[2 * i] = a - lse;
        out[2 * i + 1] = b - lse;
    }
}

static inline unsigned cdiv_u(unsigned a, unsigned b) { return (a + b - 1u) / b; }

extern "C" void kernel_launch(void* const* d_in, const int* in_sizes, int n_in,
                              void* d_out, int out_size, void* d_ws, size_t ws_size,
                              hipStream_t stream) {
    const float* x  = (const float*)d_in[0];
    const int*   ei = (const int*)d_in[1];
    const int*   row = ei;
    const int*   col = ei + N_EDGES;
    const float* w1 = (const float*)d_in[2];
    const float* b1 = (const float*)d_in[3];
    const float* w2 = (const float*)d_in[4];
    const float* b2 = (const float*)d_in[5];
    float* out = (float*)d_out;

    const size_t N = N_NODES;
    unsigned* ws = (unsigned*)d_ws;
    unsigned* deg  = ws;                         // N u32
    float*    dinv = (float*)(ws + N);           // N f32
    unsigned* B1   = ws + 2 * N;                 // N*11 (enc -> f32 in place)
    unsigned* B2   = ws + 13 * N;                // N*11
    float*    hid  = (float*)(ws + 24 * N);      // N*128
    unsigned* B3   = ws + 152 * N;               // N*128
    unsigned* B4   = ws + 280 * N;               // N*128
    float*    Wc1  = (float*)(ws + 408 * N);     // 36*128
    float*    Wc2  = Wc1 + 36 * 128;             // 384*16

    // degrees + normalization
    k_init_u32<<<cdiv_u(N_NODES, 256), 256, 0, stream>>>(deg, 0u, N_NODES);
    k_degree<<<cdiv_u(N_EDGES, 256), 256, 0, stream>>>(col, deg);
    k_dinv<<<cdiv_u(N_NODES, 256), 256, 0, stream>>>(deg, dinv);
    k_wprep<<<cdiv_u(384 * 16, 256), 256, 0, stream>>>(w1, w2, Wc1, Wc2);

    // ---- layer 1 hops (F = 11, padded thread map to 16) ----
    k_init_u32<<<cdiv_u(N_NODES * 11u, 256), 256, 0, stream>>>(B1, 0xFFFFFFFFu, N_NODES * 11u);
    k_hop<<<cdiv_u(N_EDGES * 16u, 256), 256, 0, stream>>>(x, row, col, dinv, B1, IN_F, 4);
    k_decode<<<cdiv_u(N_NODES * 11u, 256), 256, 0, stream>>>(B1, deg, IN_F, N_NODES * 11u);

    k_init_u32<<<cdiv_u(N_NODES * 11u, 256), 256, 0, stream>>>(B2, 0xFFFFFFFFu, N_NODES * 11u);
    k_hop<<<cdiv_u(N_EDGES * 16u, 256), 256, 0, stream>>>((const float*)B1, row, col, dinv, B2, IN_F, 4);
    k_decode<<<cdiv_u(N_NODES * 11u, 256), 256, 0, stream>>>(B2, deg, IN_F, N_NODES * 11u);

    // ---- layer 1 fused GEMM + bias + relu ----
    k_gemm1<<<3125, 64, 0, stream>>>(x, (const float*)B1, (const float*)B2, Wc1, b1, hid);

    // ---- layer 2 hops (F = 128) ----
    k_init_u32<<<cdiv_u(N_NODES * 128u, 256), 256, 0, stream>>>(B3, 0xFFFFFFFFu, N_NODES * 128u);
    k_hop<<<cdiv_u(N_EDGES * 128u, 256), 256, 0, stream>>>(hid, row, col, dinv, B3, HID_F, 7);
    k_decode<<<cdiv_u(N_NODES * 128u, 256), 256, 0, stream>>>(B3, deg, HID_F, N_NODES * 128u);

    k_init_u32<<<cdiv_u(N_NODES * 128u, 256), 256, 0, stream>>>(B4, 0xFFFFFFFFu, N_NODES * 128u);
    k_hop<<<cdiv_u(N_EDGES * 128u, 256), 256, 0, stream>>>((const float*)B3, row, col, dinv, B4, HID_F, 7);
    k_decode<<<cdiv_u(N_NODES * 128u, 256), 256, 0, stream>>>(B4, deg, HID_F, N_NODES * 128u);

    // ---- layer 2 GEMM + bias -> logits, then log_softmax ----
    k_gemm2<<<3125, 64, 0, stream>>>(hid, (const float*)B3, (const float*)B4, Wc2, b2, out);
    k_lsm<<<cdiv_u(N_NODES, 256), 256, 0, stream>>>(out);
}